// HydraBlock_19112604467499
// MI455X (gfx1250) — compile-verified
//
#include <hip/hip_runtime.h>
#include <hip/hip_bf16.h>
#include <math.h>

typedef __attribute__((ext_vector_type(2))) float v2f;
typedef __attribute__((ext_vector_type(4))) float f4;
typedef __attribute__((ext_vector_type(8))) float v8f;

#define D_MODEL 1024
#define D_STATE 16
#define D_CONV  4
#define D_INNER 2048
#define DT_RANK 64
#define BATCH   2
#define SEQ     2048
#define NROWS   (BATCH * SEQ)                         // 4096
#define N_IN    (2 * D_INNER + 2 * D_STATE + DT_RANK) // 4192

// ---- GEMM tiling: 256 threads = 8 waves (wave32). Wave grid 2(M) x 4(N);
// each wave computes a 64x32 tile via 4x2 fragments of V_WMMA_F32_16X16X4_F32.
// LDS tiles stored K-paired so each fragment is a single aligned ds_load_b64.
#define BM 128
#define BN 128
#define BK 16
#define APITCH (2 * BM + 4)   // floats per pair-row (+4 pad kills bank conflicts)
#define BPITCH (2 * BN + 4)

__device__ __forceinline__ float sigmoidf_(float x) { return 1.0f / (1.0f + __expf(-x)); }
__device__ __forceinline__ float siluf_(float x)    { return x * sigmoidf_(x); }
__device__ __forceinline__ float softplusf_(float x){ return (x > 20.0f) ? x : log1pf(__expf(x)); }

// C[M,N] = A[M,K] (row-major, stride lda) * W[N,K]^T (row-major weights)
// MODE 0: plain store.  MODE 1: store softplus(acc + bias[n]).
template <int MODE>
__global__ __launch_bounds__(256) void gemm_nt_wmma(
    const float* __restrict__ A, long lda,
    const float* __restrict__ W,
    const float* __restrict__ bias,
    float* __restrict__ C, long ldc,
    int M, int N, int K)
{
    // K-paired layout: element (k, m) lives at s[k>>1][2*m + (k&1)]
    __shared__ float As[BK / 2][APITCH];
    __shared__ float Bs[BK / 2][BPITCH];

    const int tid   = threadIdx.x;
    const int wave  = tid >> 5;
    const int lane  = tid & 31;
    const int half  = lane >> 4;       // ISA: lanes 0-15 hold K=0,1; lanes 16-31 hold K=2,3
    const int ml    = lane & 15;
    const int waveM = wave >> 2;       // 0..1  -> 64 rows each
    const int waveN = wave & 3;        // 0..3  -> 32 cols each
    const int blockM = blockIdx.x * BM;
    const int blockN = blockIdx.y * BN;

    // Per-thread staging coordinates (2 float4 per tile per thread)
    const int ar0 = tid >> 1;                 // A row for i=0 (0..127)
    const int ak0 = (tid & 1) << 3;           // A k-quad base: 0 or 8
    v8f acc[4][2];
#pragma unroll
    for (int i = 0; i < 4; i++)
#pragma unroll
        for (int j = 0; j < 2; j++)
#pragma unroll
            for (int r = 0; r < 8; r++) acc[i][j][r] = 0.0f;

    // Prologue: fetch first K-slab into registers
    f4 av[2], bv[2];
    {
#pragma unroll
        for (int i = 0; i < 2; i++) {
            int gm = blockM + ar0;
            int kq = ak0 + i * 4;
            av[i] = (gm < M) ? *(const f4*)(A + (long)gm * lda + kq) : (f4){0.f, 0.f, 0.f, 0.f};
            int gn = blockN + ar0;
            bv[i] = (gn < N) ? *(const f4*)(W + (long)gn * K + kq) : (f4){0.f, 0.f, 0.f, 0.f};
        }
    }

    for (int kb = 0; kb < K; kb += BK) {
        // Commit staged registers to LDS (K-paired, b64 stores)
#pragma unroll
        for (int i = 0; i < 2; i++) {
            int p = (ak0 + i * 4) >> 1;       // pair-row: 0,2,4,6 (+1)
            *(v2f*)&As[p + 0][2 * ar0] = (v2f){av[i].x, av[i].y};
            *(v2f*)&As[p + 1][2 * ar0] = (v2f){av[i].z, av[i].w};
            *(v2f*)&Bs[p + 0][2 * ar0] = (v2f){bv[i].x, bv[i].y};
            *(v2f*)&Bs[p + 1][2 * ar0] = (v2f){bv[i].z, bv[i].w};
        }
        __syncthreads();

        // Software pipeline: issue next slab's global loads before the WMMA burst
        if (kb + BK < K) {
#pragma unroll
            for (int i = 0; i < 2; i++) {
                int gm = blockM + ar0;
                int kq = kb + BK + ak0 + i * 4;
                av[i] = (gm < M) ? *(const f4*)(A + (long)gm * lda + kq) : (f4){0.f, 0.f, 0.f, 0.f};
                int gn = blockN + ar0;
                bv[i] = (gn < N) ? *(const f4*)(W + (long)gn * K + kq) : (f4){0.f, 0.f, 0.f, 0.f};
            }
        }

#pragma unroll
        for (int kk = 0; kk < BK; kk += 4) {
            const int p = (kk >> 1) + half;   // lanes 16-31 take the next K pair
            v2f a[4], b[2];
#pragma unroll
            for (int mi = 0; mi < 4; mi++)
                a[mi] = *(const v2f*)&As[p][2 * (waveM * 64 + mi * 16 + ml)];
#pragma unroll
            for (int ni = 0; ni < 2; ni++)
                b[ni] = *(const v2f*)&Bs[p][2 * (waveN * 32 + ni * 16 + ml)];
#pragma unroll
            for (int mi = 0; mi < 4; mi++)
#pragma unroll
                for (int ni = 0; ni < 2; ni++)
                    acc[mi][ni] = __builtin_amdgcn_wmma_f32_16x16x4_f32(
                        false, a[mi], false, b[ni], (short)0, acc[mi][ni], false, false);
        }
        __syncthreads();
    }

    // Epilogue: C/D layout — VGPR r holds M=r (lanes 0-15) / M=r+8 (lanes 16-31), N=lane&15
#pragma unroll
    for (int mi = 0; mi < 4; mi++) {
#pragma unroll
        for (int ni = 0; ni < 2; ni++) {
#pragma unroll
            for (int r = 0; r < 8; r++) {
                int m = blockM + waveM * 64 + mi * 16 + r + 8 * half;
                int n = blockN + waveN * 32 + ni * 16 + ml;
                if (m < M && n < N) {
                    float v = acc[mi][ni][r];
                    if (MODE == 1) v = softplusf_(v + bias[n]);
                    C[(long)m * ldc + n] = v;
                }
            }
        }
    }
}

// Depthwise causal conv1d (width 4, left pad 3) + bias + SiLU over x slice of zxbcdt
__global__ __launch_bounds__(256) void conv_silu_kernel(
    const float* __restrict__ zxbcdt,
    const float* __restrict__ cw,
    const float* __restrict__ cb,
    float* __restrict__ xconv)
{
    long idx = (long)blockIdx.x * 256 + threadIdx.x;
    if (idx >= (long)NROWS * D_INNER) return;
    int  d    = (int)(idx & (D_INNER - 1));
    long row  = idx >> 11;            // b*SEQ + s
    int  s    = (int)(row & (SEQ - 1));
    long brow = row - s;              // b*SEQ
    float acc = cb[d];
#pragma unroll
    for (int j = 0; j < D_CONV; j++) {
        int sj = s - (D_CONV - 1) + j;
        float xv = (sj >= 0) ? zxbcdt[(brow + sj) * (long)N_IN + D_INNER + d] : 0.0f;
        acc = fmaf(xv, cw[d * D_CONV + j], acc);
    }
    xconv[idx] = siluf_(acc);
}

// Selective-scan: one thread per (b, d_inner) channel, 16 states in registers.
// dir=0: forward, plain store of (scan + D*x). dir=1: backward, accumulate.
__global__ __launch_bounds__(256) void ssm_scan_kernel(
    const float* __restrict__ zxbcdt,
    const float* __restrict__ xconv,
    const float* __restrict__ dtbuf,
    const float* __restrict__ A_log,
    const float* __restrict__ D_param,
    float* __restrict__ ybuf,
    int dir)
{
    __shared__ float sB[D_STATE], sC[D_STATE];
    int t = blockIdx.x * 256 + threadIdx.x;   // 0..4095; blocks never span b
    int b = t >> 11;
    int d = t & (D_INNER - 1);

    float Ac[D_STATE], st[D_STATE];
#pragma unroll
    for (int n = 0; n < D_STATE; n++) {
        Ac[n] = -__expf(A_log[d * D_STATE + n]);
        st[n] = 0.0f;
    }
    float Dp = D_param[d];

    for (int i = 0; i < SEQ; i++) {
        int  s   = dir ? (SEQ - 1 - i) : i;
        long row = (long)b * SEQ + s;
        __syncthreads();
        if (threadIdx.x < D_STATE)
            sB[threadIdx.x] = zxbcdt[row * N_IN + 2 * D_INNER + threadIdx.x];
        else if (threadIdx.x < 2 * D_STATE)
            sC[threadIdx.x - D_STATE] = zxbcdt[row * N_IN + 2 * D_INNER + threadIdx.x];
        __syncthreads();

        float x  = xconv[row * D_INNER + d];
        float dt = dtbuf[row * D_INNER + d];
        float y  = Dp * x;
#pragma unroll
        for (int n = 0; n < D_STATE; n++) {
            st[n] = fmaf(dt * Ac[n], st[n], st[n]) + x * sB[n];   // (1+dt*A)*st + x*B
            y     = fmaf(st[n], sC[n], y);
        }
        long o = row * D_INNER + d;
        if (dir) ybuf[o] += y; else ybuf[o] = y;
    }
}

// RMSNorm over d_inner + SiLU(z) gate, in place on ybuf. One block per (b,s) row.
__global__ __launch_bounds__(256) void rmsnorm_gate_kernel(
    const float* __restrict__ zxbcdt,
    const float* __restrict__ norm_w,
    float* __restrict__ ybuf)
{
    __shared__ float red[256];
    long row = blockIdx.x;
    int  tid = threadIdx.x;
    float vals[8];
    float ss = 0.0f;
#pragma unroll
    for (int i = 0; i < 8; i++) {
        int d = tid + i * 256;
        float v = ybuf[row * D_INNER + d];
        vals[i] = v;
        ss = fmaf(v, v, ss);
    }
    red[tid] = ss;
    __syncthreads();
    for (int off = 128; off > 0; off >>= 1) {
        if (tid < off) red[tid] += red[tid + off];
        __syncthreads();
    }
    float rinv = rsqrtf(red[0] * (1.0f / D_INNER) + 1e-5f);
#pragma unroll
    for (int i = 0; i < 8; i++) {
        int d = tid + i * 256;
        float z = zxbcdt[row * N_IN + d];
        ybuf[row * D_INNER + d] = norm_w[d] * vals[i] * rinv * siluf_(z);
    }
}

extern "C" void kernel_launch(void* const* d_in, const int* in_sizes, int n_in,
                              void* d_out, int out_size, void* d_ws, size_t ws_size,
                              hipStream_t stream)
{
    const float* u       = (const float*)d_in[0];
    const float* W_in    = (const float*)d_in[1];
    const float* conv_w  = (const float*)d_in[2];
    const float* conv_b  = (const float*)d_in[3];
    const float* A_log   = (const float*)d_in[4];
    const float* D_param = (const float*)d_in[5];
    const float* W_dt    = (const float*)d_in[6];
    const float* b_dt    = (const float*)d_in[7];
    const float* norm_w  = (const float*)d_in[8];
    const float* W_out   = (const float*)d_in[9];
    float* out = (float*)d_out;

    float* ws     = (float*)d_ws;
    float* zxbcdt = ws;                                  // 4096 * 4192
    float* xconv  = zxbcdt + (long)NROWS * N_IN;         // 4096 * 2048
    float* dtbuf  = xconv  + (long)NROWS * D_INNER;      // 4096 * 2048
    float* ybuf   = dtbuf  + (long)NROWS * D_INNER;      // 4096 * 2048

    dim3 blk(256);

    // 1) in_proj: zxbcdt = u @ W_in^T        (4096 x 1024 x 4192)
    gemm_nt_wmma<0><<<dim3(NROWS / BM, (N_IN + BN - 1) / BN), blk, 0, stream>>>(
        u, D_MODEL, W_in, nullptr, zxbcdt, N_IN, NROWS, N_IN, D_MODEL);

    // 2) depthwise causal conv + SiLU on x slice
    long nconv = (long)NROWS * D_INNER;
    conv_silu_kernel<<<dim3((unsigned)((nconv + 255) / 256)), blk, 0, stream>>>(
        zxbcdt, conv_w, conv_b, xconv);

    // 3) dt = softplus(dt_raw @ W_dt^T + b_dt)   (4096 x 64 x 2048)
    gemm_nt_wmma<1><<<dim3(NROWS / BM, D_INNER / BN), blk, 0, stream>>>(
        zxbcdt + 2 * D_INNER + 2 * D_STATE, N_IN, W_dt, b_dt, dtbuf, D_INNER,
        NROWS, D_INNER, DT_RANK);

    // 4) forward scan (store), 5) backward scan (accumulate)
    ssm_scan_kernel<<<dim3(16), blk, 0, stream>>>(
        zxbcdt, xconv, dtbuf, A_log, D_param, ybuf, 0);
    ssm_scan_kernel<<<dim3(16), blk, 0, stream>>>(
        zxbcdt, xconv, dtbuf, A_log, D_param, ybuf, 1);

    // 6) RMSNorm + SiLU(z) gate
    rmsnorm_gate_kernel<<<dim3(NROWS), blk, 0, stream>>>(zxbcdt, norm_w, ybuf);

    // 7) out_proj: out = y @ W_out^T         (4096 x 2048 x 1024)
    gemm_nt_wmma<0><<<dim3(NROWS / BM, D_MODEL / BN), blk, 0, stream>>>(
        ybuf, D_INNER, W_out, nullptr, out, D_MODEL, NROWS, D_MODEL, D_INNER);
}